// CapsuleLayer_82463372083896
// MI455X (gfx1250) — compile-verified
//
#include <hip/hip_runtime.h>
#include <math.h>

// ---------------------------------------------------------------------------
// Capsule routing with u_hat (188.7 MB) never materialized; both routing
// contractions are fp32 WMMA GEMMs (V_WMMA_F32_16X16X4_F32), N-register-
// blocked: each wave owns one 16-row M tile x all 10 N tiles (N=160), so A is
// loaded once per k-step and reused across 10 back-to-back WMMAs.
//
//   B = 256, in_units U = 8, in_size I = 1152, out_units J = 10, out_size D = 16
//   GEMM1: S[256,160]  = X[256,9216] * Wc[9216,160]   (K = u*1152+i, contiguous in x)
//   GEMM2: G[9216,160] = X^T * V[256,160]
// ---------------------------------------------------------------------------

typedef __attribute__((ext_vector_type(2))) float v2f;
typedef __attribute__((ext_vector_type(8))) float v8f;

#define BATCH 256
#define IN_U 8
#define IN_S 1152
#define OUT_J 10
#define OUT_D 16
#define NDIM 160          /* OUT_J*OUT_D  */
#define KDIM 9216         /* IN_U*IN_S    */
#define NT 10             /* N tiles of 16 per wave */

// ---------------- tiny helper kernels ----------------

__global__ void init_b_kernel(float* bij) {
    int idx = blockIdx.x * blockDim.x + threadIdx.x;
    if (idx < IN_S * OUT_J) bij[idx] = 1.0f;
}

// softmax over in_size (axis 0) for each of the 10 columns; one block per j
__global__ __launch_bounds__(256) void softmax_kernel(const float* __restrict__ bij,
                                                      float* __restrict__ cij) {
    __shared__ float red[256];
    const int j = blockIdx.x;
    const int t = threadIdx.x;
    float mx = -INFINITY;
    for (int i = t; i < IN_S; i += 256) mx = fmaxf(mx, bij[i * OUT_J + j]);
    red[t] = mx; __syncthreads();
    for (int s = 128; s > 0; s >>= 1) {
        if (t < s) red[t] = fmaxf(red[t], red[t + s]);
        __syncthreads();
    }
    mx = red[0]; __syncthreads();
    float sum = 0.0f;
    for (int i = t; i < IN_S; i += 256) sum += expf(bij[i * OUT_J + j] - mx);
    red[t] = sum; __syncthreads();
    for (int s = 128; s > 0; s >>= 1) {
        if (t < s) red[t] += red[t + s];
        __syncthreads();
    }
    const float inv = 1.0f / red[0];
    for (int i = t; i < IN_S; i += 256)
        cij[i * OUT_J + j] = expf(bij[i * OUT_J + j] - mx) * inv;
}

// Wc[(u*IN_S+i)*160 + j*16+d] = W[i*1280 + j*128 + d*8 + u] * c[i*10+j]
__global__ __launch_bounds__(256) void prep_wc_kernel(const float* __restrict__ W,
                                                      const float* __restrict__ cij,
                                                      float* __restrict__ Wc) {
    int idx = blockIdx.x * blockDim.x + threadIdx.x;
    if (idx >= KDIM * NDIM) return;
    int m = idx / NDIM;              // u*IN_S + i
    int n = idx - m * NDIM;          // j*16 + d
    int u = m / IN_S;
    int i = m - u * IN_S;
    int j = n >> 4;
    int d = n & 15;
    Wc[idx] = W[i * (OUT_J * OUT_D * IN_U) + j * (OUT_D * IN_U) + d * IN_U + u]
              * cij[i * OUT_J + j];
}

// ---------------- WMMA GEMM 1: S[256,160] = X[256,9216] * Wc[9216,160] ------
// grid (2), block 256 (8 waves). Wave w: M tile (blockIdx.x*8+w)*16, all 10 N
// tiles register-blocked; A pair loaded once per k-step, reused by 10 WMMAs.
__global__ __launch_bounds__(256) void gemm_s_kernel(const float* __restrict__ X,
                                                     const float* __restrict__ Wc,
                                                     float* __restrict__ S) {
    const int wave = threadIdx.x >> 5;
    const int lane = threadIdx.x & 31;
    const int half = lane >> 4;      // K/M pair selector per ISA VGPR layout
    const int lm   = lane & 15;
    const int m0 = (blockIdx.x * 8 + wave) * 16;

    v8f acc[NT];
#pragma unroll
    for (int nt = 0; nt < NT; ++nt) acc[nt] = (v8f){};

    const float* arow = X + (size_t)(m0 + lm) * KDIM + 2 * half;
#pragma unroll 2
    for (int k0 = 0; k0 < KDIM; k0 += 4) {
        const v2f a = *(const v2f*)(arow + k0);        // A[lm][k0+2h .. +1]
        const float* bp = Wc + (size_t)(k0 + 2 * half) * NDIM + lm;
#pragma unroll
        for (int nt = 0; nt < NT; ++nt) {
            v2f bm;
            bm.x = bp[nt * 16];
            bm.y = bp[NDIM + nt * 16];
            acc[nt] = __builtin_amdgcn_wmma_f32_16x16x4_f32(
                false, a, false, bm, (short)0, acc[nt], false, false);
        }
    }
#pragma unroll
    for (int nt = 0; nt < NT; ++nt)
#pragma unroll
        for (int r = 0; r < 8; ++r)
            S[(size_t)(m0 + r + 8 * half) * NDIM + nt * 16 + lm] = acc[nt][r];
}

// ---------------- WMMA GEMM 2: G[9216,160] = X^T * V, K = batch = 256 -------
// grid (72), block 256 (8 waves); same N-register-blocking.
__global__ __launch_bounds__(256) void gemm_g_kernel(const float* __restrict__ X,
                                                     const float* __restrict__ V,
                                                     float* __restrict__ G) {
    const int wave = threadIdx.x >> 5;
    const int lane = threadIdx.x & 31;
    const int half = lane >> 4;
    const int lm   = lane & 15;
    const int m0 = (blockIdx.x * 8 + wave) * 16;

    v8f acc[NT];
#pragma unroll
    for (int nt = 0; nt < NT; ++nt) acc[nt] = (v8f){};

#pragma unroll 2
    for (int k0 = 0; k0 < BATCH; k0 += 4) {
        const int ka = k0 + 2 * half;
        v2f a;
        a.x = X[(size_t)ka * KDIM + m0 + lm];          // A[m][k] = X[k][m]
        a.y = X[(size_t)(ka + 1) * KDIM + m0 + lm];
        const float* bp = V + (size_t)ka * NDIM + lm;
#pragma unroll
        for (int nt = 0; nt < NT; ++nt) {
            v2f bm;
            bm.x = bp[nt * 16];
            bm.y = bp[NDIM + nt * 16];
            acc[nt] = __builtin_amdgcn_wmma_f32_16x16x4_f32(
                false, a, false, bm, (short)0, acc[nt], false, false);
        }
    }
#pragma unroll
    for (int nt = 0; nt < NT; ++nt)
#pragma unroll
        for (int r = 0; r < 8; ++r)
            G[(size_t)(m0 + r + 8 * half) * NDIM + nt * 16 + lm] = acc[nt][r];
}

// squash: norm over out_units axis j (per reference). One block per batch b.
__global__ __launch_bounds__(160) void squash_kernel(const float* __restrict__ S,
                                                     float* __restrict__ V) {
    __shared__ float sh[NDIM];
    const int b = blockIdx.x;
    const int t = threadIdx.x;       // t = j*16 + d
    sh[t] = S[b * NDIM + t];
    __syncthreads();
    const int d = t & 15;
    float msq = 0.0f;
#pragma unroll
    for (int j = 0; j < OUT_J; ++j) {
        float xv = sh[j * 16 + d];
        msq += xv * xv;
    }
    const float mag = sqrtf(msq);
    V[b * NDIM + t] = msq / (1.0f + msq) * (sh[t] / mag);
}

// b_ij[i,j] += (1/B) * sum_{d,u} W[i,j,d,u] * G[(u*IN_S+i)*160 + j*16+d]
__global__ __launch_bounds__(256) void update_b_kernel(const float* __restrict__ W,
                                                       const float* __restrict__ G,
                                                       float* __restrict__ bij) {
    int idx = blockIdx.x * blockDim.x + threadIdx.x;
    if (idx >= IN_S * OUT_J) return;
    int i = idx / OUT_J;
    int j = idx - i * OUT_J;
    float acc = 0.0f;
    const float* wp = W + i * (OUT_J * OUT_D * IN_U) + j * (OUT_D * IN_U);
#pragma unroll 4
    for (int d = 0; d < OUT_D; ++d) {
        const float* gp = G + (size_t)i * NDIM + j * 16 + d;
#pragma unroll
        for (int u = 0; u < IN_U; ++u)
            acc += wp[d * IN_U + u] * gp[(size_t)u * IN_S * NDIM];
    }
    bij[idx] += acc * (1.0f / (float)BATCH);
}

// ---------------------------------------------------------------------------

extern "C" void kernel_launch(void* const* d_in, const int* in_sizes, int n_in,
                              void* d_out, int out_size, void* d_ws, size_t ws_size,
                              hipStream_t stream) {
    (void)in_sizes; (void)n_in; (void)out_size; (void)ws_size;
    const float* x = (const float*)d_in[0];   // (256, 8, 1152) == (B, K) flat
    const float* W = (const float*)d_in[1];   // (1152, 10, 16, 8)
    float* out = (float*)d_out;               // (256, 10, 16, 1)

    // workspace layout (floats); Wc and G lifetimes are disjoint -> share "big"
    float* ws  = (float*)d_ws;
    float* bij = ws;                              // 11520
    float* cij = bij + IN_S * OUT_J;              // 11520
    float* S   = cij + IN_S * OUT_J;              // 40960
    float* V   = S + BATCH * NDIM;                // 40960
    float* big = V + BATCH * NDIM;                // 1,474,560 (Wc, then G)

    init_b_kernel<<<dim3(45), dim3(256), 0, stream>>>(bij);

    const int NUM_ITERATIONS = 3;
    for (int it = 0; it < NUM_ITERATIONS; ++it) {
        softmax_kernel<<<dim3(OUT_J), dim3(256), 0, stream>>>(bij, cij);
        prep_wc_kernel<<<dim3((KDIM * NDIM + 255) / 256), dim3(256), 0, stream>>>(W, cij, big);
        gemm_s_kernel<<<dim3(2), dim3(256), 0, stream>>>(x, big, S);
        float* vdst = (it == NUM_ITERATIONS - 1) ? out : V;
        squash_kernel<<<dim3(BATCH), dim3(NDIM), 0, stream>>>(S, vdst);
        if (it < NUM_ITERATIONS - 1) {
            gemm_g_kernel<<<dim3(72), dim3(256), 0, stream>>>(x, V, big);
            update_b_kernel<<<dim3(45), dim3(256), 0, stream>>>(W, big, bij);
        }
    }
}